// APPNP_78709570667357
// MI455X (gfx1250) — compile-verified
//
#include <hip/hip_runtime.h>
#include <cstdint>
#include <cstddef>

// Problem constants (match reference)
#define NN      50000
#define D_INF   512
#define D_HIDF  128
#define N_CLSF  64
#define K_ITERS 10
#define ALPHA_F 0.1f

typedef __attribute__((ext_vector_type(2))) float v2f;
typedef __attribute__((ext_vector_type(4))) float v4f;
typedef __attribute__((ext_vector_type(8))) float v8f;

static inline int cdiv(long long a, long long b) { return (int)((a + b - 1) / b); }

// ---------------------------------------------------------------- utilities
__global__ void zero_f32_v4(v4f* __restrict__ p, long long nq) {
  long long i = (long long)blockIdx.x * blockDim.x + threadIdx.x;
  if (i < nq) p[i] = (v4f){0.0f, 0.0f, 0.0f, 0.0f};
}

__global__ void zero_f32(float* __restrict__ p, long long n) {
  long long i = (long long)blockIdx.x * blockDim.x + threadIdx.x;
  if (i < n) p[i] = 0.0f;
}

__global__ void degree_count(const int* __restrict__ dst, float* __restrict__ deg, int e) {
  int i = blockIdx.x * blockDim.x + threadIdx.x;
  if (i < e) atomicAdd(&deg[dst[i]], 1.0f);
}

// deg -> dinv in place; +1 accounts for the self-loop, so deg >= 1 always.
__global__ void deg_to_dinv(float* __restrict__ deg, int n) {
  int i = blockIdx.x * blockDim.x + threadIdx.x;
  if (i < n) deg[i] = rsqrtf(deg[i] + 1.0f);
}

__global__ void edge_norm(const int* __restrict__ src, const int* __restrict__ dst,
                          const float* __restrict__ dinv, float* __restrict__ nrm, int e) {
  int i = blockIdx.x * blockDim.x + threadIdx.x;
  if (i < e) nrm[i] = dinv[src[i]] * dinv[dst[i]];
}

// ------------------------------------------------------- WMMA f32 GEMM(+bias[+relu])
// C[M,NOUT] = act(A[M,KDIM] @ W[NOUT,KDIM]^T + bias).  One wave per 16-row tile,
// computing the full 16 x NOUT strip with NOUT/16 accumulators. Requires M % 16 == 0
// (50000 = 3125*16) so EXEC is all-ones in every wave that reaches the WMMA.
//
// V_WMMA_F32_16X16X4_F32 operand layout (ISA 7.12.2):
//   A 16x4:  lane L, vgpr v -> A[L%16][2*(L/16)+v]   (contiguous pair -> one b64 load)
//   B 4x16:  lane L, vgpr v -> B[2*(L/16)+v][L%16] = W[L%16][k] (W row-major, k-contig)
//   D 16x16: lane L, vgpr r -> D[r + 8*(L/16)][L%16]
template <int KDIM, int NOUT, bool RELU>
__global__ __launch_bounds__(256) void gemm_wmma_f32(const float* __restrict__ A,
                                                     const float* __restrict__ W,
                                                     const float* __restrict__ bias,
                                                     float* __restrict__ C, int M) {
  constexpr int NT = NOUT / 16;
  const int wave = blockIdx.x * (blockDim.x >> 5) + (threadIdx.x >> 5);
  const int m0 = wave * 16;
  if (m0 >= M) return;  // wave-uniform: EXEC stays all-1s for active waves

  const int lane = threadIdx.x & 31;
  const int lm = lane & 15;   // row-in-tile (A) / col-in-tile (B,D)
  const int kg = lane >> 4;   // K-group 0/1

  v8f acc[NT] = {};

  const float* arow = A + (size_t)(m0 + lm) * KDIM + 2 * kg;
  for (int k4 = 0; k4 < KDIM; k4 += 4) {
    v2f a = *(const v2f*)(arow + k4);
#pragma unroll
    for (int t = 0; t < NT; ++t) {
      v2f b = *(const v2f*)(W + (size_t)(t * 16 + lm) * KDIM + k4 + 2 * kg);
      acc[t] = __builtin_amdgcn_wmma_f32_16x16x4_f32(
          /*neg_a=*/false, a, /*neg_b=*/false, b,
          /*c_mod=*/(short)0, acc[t], /*reuse_a=*/false, /*reuse_b=*/false);
    }
  }

#pragma unroll
  for (int t = 0; t < NT; ++t) {
    const int n = t * 16 + lm;
    const float bz = bias[n];
#pragma unroll
    for (int r = 0; r < 8; ++r) {
      float v = acc[t][r] + bz;
      if (RELU) v = v > 0.0f ? v : 0.0f;
      C[(size_t)(m0 + kg * 8 + r) * NOUT + n] = v;
    }
  }
}

// ------------------------------------------------------- APPNP diffusion step
// One thread per (edge, 4 features): edge scalars (src/dst/nrm) amortized 4x,
// gather becomes a b128 load, and the 4 atomic f32 adds hit consecutive dwords
// of one cacheline. h1/agg/norm all live in the 192MB L2 -> L2-bandwidth bound.
template <int LOGD>
__global__ void scatter_add4(const float* __restrict__ cur, const int* __restrict__ src,
                             const int* __restrict__ dst, const float* __restrict__ nrm,
                             float* __restrict__ agg, int e) {
  constexpr int LOGQ = LOGD - 2;                 // quads per edge row
  long long i = (long long)blockIdx.x * blockDim.x + threadIdx.x;
  long long total = (long long)e << LOGQ;
  if (i >= total) return;
  const int ed = (int)(i >> LOGQ);
  const int f0 = (int)((i & ((1 << LOGQ) - 1)) << 2);
  const int s = src[ed];
  const int d = dst[ed];
  const float w = nrm[ed];
  const v4f c = *(const v4f*)(cur + ((size_t)s << LOGD) + f0);
  float* a = agg + ((size_t)d << LOGD) + f0;
  atomicAdd(a + 0, c.x * w);
  atomicAdd(a + 1, c.y * w);
  atomicAdd(a + 2, c.z * w);
  atomicAdd(a + 3, c.w * w);
}

// cur = (1-a) * (agg + dinv[v]^2 * cur) + a * h0   (self-loop folded in here)
template <int LOGD>
__global__ void combine_step4(const v4f* __restrict__ agg, const v4f* __restrict__ h0,
                              const float* __restrict__ dinv, v4f* __restrict__ cur,
                              int n) {
  constexpr int LOGQ = LOGD - 2;
  long long i = (long long)blockIdx.x * blockDim.x + threadIdx.x;
  long long total = (long long)n << LOGQ;
  if (i >= total) return;
  const int v = (int)(i >> LOGQ);
  const float di = dinv[v];
  const float sl = di * di;
  const v4f ag = agg[i], cu = cur[i], hz = h0[i];
  v4f r;
  r.x = (1.0f - ALPHA_F) * (ag.x + sl * cu.x) + ALPHA_F * hz.x;
  r.y = (1.0f - ALPHA_F) * (ag.y + sl * cu.y) + ALPHA_F * hz.y;
  r.z = (1.0f - ALPHA_F) * (ag.z + sl * cu.z) + ALPHA_F * hz.z;
  r.w = (1.0f - ALPHA_F) * (ag.w + sl * cu.w) + ALPHA_F * hz.w;
  cur[i] = r;
}

// ---------------------------------------------------------------- host driver
extern "C" void kernel_launch(void* const* d_in, const int* in_sizes, int n_in,
                              void* d_out, int out_size, void* d_ws, size_t ws_size,
                              hipStream_t stream) {
  const float* x  = (const float*)d_in[0];
  const int*   ei = (const int*)d_in[1];
  const float* W1 = (const float*)d_in[2];
  const float* b1 = (const float*)d_in[3];
  const float* W2 = (const float*)d_in[4];
  const float* b2 = (const float*)d_in[5];

  const int N = in_sizes[0] / D_INF;   // 50000
  const int E = in_sizes[1] / 2;       // 800000
  const int* src = ei;       // edge_index[0] = message sources
  const int* dst = ei + E;   // edge_index[1] = aggregation targets

  float* out     = (float*)d_out;
  float* out_x   = out;                                   // [N, 512]
  float* out_emb = out_x + (size_t)N * D_INF;             // [N, 128]
  float* out_lg  = out_emb + (size_t)N * D_HIDF;          // [N, 64]

  // workspace layout (~93 MB total; every segment is 16B-aligned)
  char* w = (char*)d_ws;
  float* dinv = (float*)w; w += sizeof(float) * (size_t)N;            // deg -> dinv
  float* nrm  = (float*)w; w += sizeof(float) * (size_t)E;
  float* h1   = (float*)w; w += sizeof(float) * (size_t)N * D_HIDF;   // h0 for emb
  float* cur  = (float*)w; w += sizeof(float) * (size_t)N * D_HIDF;   // diffusion state
  float* agg  = (float*)w; w += sizeof(float) * (size_t)N * D_HIDF;   // segment-sum target
  float* l0   = (float*)w; w += sizeof(float) * (size_t)N * N_CLSF;   // h0 for logits

  const int T = 256;

  // ---- GCN normalization: deg -> dinv -> per-edge norm
  zero_f32<<<cdiv(N, T), T, 0, stream>>>(dinv, N);
  degree_count<<<cdiv(E, T), T, 0, stream>>>(dst, dinv, E);
  deg_to_dinv<<<cdiv(N, T), T, 0, stream>>>(dinv, N);
  edge_norm<<<cdiv(E, T), T, 0, stream>>>(src, dst, dinv, nrm, E);

  // ---- h1 = relu(x @ W1^T + b1)   (WMMA f32, one wave per 16-row tile)
  {
    const int tiles = N / 16;                 // 3125
    const int blocks = cdiv(tiles, T / 32);   // 8 waves/block
    gemm_wmma_f32<D_INF, D_HIDF, true><<<blocks, T, 0, stream>>>(x, W1, b1, h1, N);
  }

  // ---- emb1 = APPNP(h1), D = 128
  hipMemcpyAsync(cur, h1, sizeof(float) * (size_t)N * D_HIDF,
                 hipMemcpyDeviceToDevice, stream);
  {
    const long long ndq = ((long long)N * D_HIDF) >> 2;  // quads
    const long long edq = ((long long)E * D_HIDF) >> 2;
    for (int k = 0; k < K_ITERS; ++k) {
      zero_f32_v4<<<cdiv(ndq, T), T, 0, stream>>>((v4f*)agg, ndq);
      scatter_add4<7><<<cdiv(edq, T), T, 0, stream>>>(cur, src, dst, nrm, agg, E);
      combine_step4<7><<<cdiv(ndq, T), T, 0, stream>>>((const v4f*)agg, (const v4f*)h1,
                                                       dinv, (v4f*)cur, N);
    }
    hipMemcpyAsync(out_emb, cur, sizeof(float) * (size_t)N * D_HIDF,
                   hipMemcpyDeviceToDevice, stream);
  }

  // ---- logits = h1 @ W2^T + b2   (note: input is pre-diffusion h1)
  {
    const int tiles = N / 16;
    const int blocks = cdiv(tiles, T / 32);
    gemm_wmma_f32<D_HIDF, N_CLSF, false><<<blocks, T, 0, stream>>>(h1, W2, b2, l0, N);
  }

  // ---- out = APPNP(logits), D = 64 (reuse cur/agg buffers)
  hipMemcpyAsync(cur, l0, sizeof(float) * (size_t)N * N_CLSF,
                 hipMemcpyDeviceToDevice, stream);
  {
    const long long ndq = ((long long)N * N_CLSF) >> 2;
    const long long edq = ((long long)E * N_CLSF) >> 2;
    for (int k = 0; k < K_ITERS; ++k) {
      zero_f32_v4<<<cdiv(ndq, T), T, 0, stream>>>((v4f*)agg, ndq);
      scatter_add4<6><<<cdiv(edq, T), T, 0, stream>>>(cur, src, dst, nrm, agg, E);
      combine_step4<6><<<cdiv(ndq, T), T, 0, stream>>>((const v4f*)agg, (const v4f*)l0,
                                                       dinv, (v4f*)cur, N);
    }
    hipMemcpyAsync(out_lg, cur, sizeof(float) * (size_t)N * N_CLSF,
                   hipMemcpyDeviceToDevice, stream);
  }

  // ---- passthrough x
  hipMemcpyAsync(out_x, x, sizeof(float) * (size_t)N * D_INF,
                 hipMemcpyDeviceToDevice, stream);
}